// MultiHeadAttention_77343771066853
// MI455X (gfx1250) — compile-verified
//
#include <hip/hip_runtime.h>
#include <cmath>

// ---------------------------------------------------------------------------
// MI455X (gfx1250) flash-decode GQA with paged KV cache.
// Roofline: 256 MiB KV stream dominates (≈11.5us @ 23.3 TB/s); FLOPs (<1 GFLOP)
// are negligible. K/V tiles are DMA'd into LDS by the Tensor Data Mover
// (TENSOR_LOAD_TO_LDS, TENSORcnt) with hardware padding (stride 64+4 dwords)
// for bank-conflict-free WMMA fragment reads; matrix math uses
// V_WMMA_F32_16X16X4_F32 (fp32-native, no conversion overhead).
// ---------------------------------------------------------------------------

typedef __attribute__((ext_vector_type(2))) float v2f;
typedef __attribute__((ext_vector_type(8))) float v8f;
typedef __attribute__((ext_vector_type(4))) unsigned int v4u;
typedef __attribute__((ext_vector_type(8))) int v8i;
typedef __attribute__((ext_vector_type(4))) int v4i;

constexpr int BATCH = 16, NBLK = 16, TBLK = 256;
constexpr int EMBED = 2048, NKV = 8, HD = 64, G = 4;   // NH=32, g = NH/NKV = 4
constexpr int BHN = BATCH * NKV;                        // 128 (b, kv-head) pairs
constexpr float SCALE = 0.125f;                         // 1/sqrt(64)

#if __has_builtin(__builtin_amdgcn_tensor_load_to_lds)
#define USE_TDM 1
#else
#define USE_TDM 0
#endif

__device__ __forceinline__ v8f wmma_f32(v2f a, v2f b, v8f c) {
  // D = A(16x4 f32) * B(4x16 f32) + C(16x16 f32)
  return __builtin_amdgcn_wmma_f32_16x16x4_f32(false, a, false, b,
                                               (short)0, c, false, false);
}

#if USE_TDM
// Issue one TDM load: 2D tile rows x 64 f32, global row stride 64 f32,
// LDS destination padded +4 dwords after every 64 dwords (row stride 68 f32).
__device__ __forceinline__ void tdm_load_tile(const float* gsrc, float* lds_dst,
                                              int rows) {
  const unsigned long long ga = (unsigned long long)(const void*)gsrc;
  const unsigned lds_addr = (unsigned)(uintptr_t)(void*)lds_dst;
  v4u g0;
  g0[0] = 1u;                                        // count=1, user mode
  g0[1] = lds_addr;                                  // lds_addr
  g0[2] = (unsigned)(ga & 0xffffffffu);              // global_addr[31:0]
  g0[3] = ((unsigned)(ga >> 32) & 0x01ffffffu) | (2u << 30);  // addr[56:32]|type=2
  v8i g1;
  // wg_mask=0 | data_size=4B | pad_enable | pad_interval=64dw | pad_amount=4dw
  g1[0] = (int)((2u << 16) | (1u << 20) | (5u << 22) | (3u << 25));
  g1[1] = (int)(64u << 16);          // tensor_dim0 = 64  (bits 79:48)
  g1[2] = (int)(256u << 16);         // tensor_dim1 = 256 (bits 111:80)
  g1[3] = (int)(64u << 16);          // tile_dim0 = 64    (bits 127:112)
  g1[4] = rows;                      // tile_dim1 = rows, tile_dim2 = 0
  g1[5] = 64;                        // tensor_dim0_stride = 64
  g1[6] = 0;                         // dim0_stride hi / dim1_stride lo
  g1[7] = 0;                         // tensor_dim1_stride hi (unused, 2D)
  v4i gz = {};                       // groups 2/3: zero (tile_dim2==0 -> 2D)
#if defined(__clang_major__) && __clang_major__ >= 23
  v8i gz8 = {};
  __builtin_amdgcn_tensor_load_to_lds(g0, g1, gz, gz, gz8, 0);
#else
  __builtin_amdgcn_tensor_load_to_lds(g0, g1, gz, gz, 0);
#endif
}

__device__ __forceinline__ void tdm_wait() {
#if __has_builtin(__builtin_amdgcn_s_wait_tensorcnt)
  __builtin_amdgcn_s_wait_tensorcnt(0);
#else
  asm volatile("s_wait_tensorcnt 0x0" ::: "memory");
#endif
}
#endif  // USE_TDM

// ---------------------------------------------------------------------------
// Y[r][o] = sum_k X[r][k] * W[o][k]     X: 16 x 2048, W: Nout x 2048
// One wave per 16(batch) x 16(out) tile; X staged once in padded LDS
// (stride 2052 floats -> WMMA A-fragment reads are bank-conflict free).
// ---------------------------------------------------------------------------
__global__ __launch_bounds__(256) void gemm16_xWt(const float* __restrict__ X,
                                                  const float* __restrict__ W,
                                                  float* __restrict__ Y,
                                                  int Nout) {
  __shared__ float Xs[16 * 2052];
  const int tid = threadIdx.x;
  for (int i = tid; i < 16 * 512; i += 256) {          // 512 float4 per row
    const int row = i >> 9, c4 = i & 511;
    const float4 v = ((const float4*)X)[row * 512 + c4];
    *(float4*)&Xs[row * 2052 + (c4 << 2)] = v;
  }
  __syncthreads();

  const int wave = tid >> 5, lane = tid & 31;
  const int n0 = (blockIdx.x * 8 + wave) * 16;         // output-feature tile
  if (n0 >= Nout) return;                              // wave-uniform
  const int m = lane & 15;
  const int koff = (lane < 16) ? 0 : 2;                // K-half per A/B layout
  const float* wrow = W + (size_t)(n0 + m) * EMBED + koff;
  const float* xrow = Xs + m * 2052 + koff;

  v8f c = {};
  for (int k0 = 0; k0 < EMBED; k0 += 4) {
    v2f a, b;
    a.x = xrow[k0];
    a.y = xrow[k0 + 1];
    const float2 wv = *(const float2*)(wrow + k0);
    b.x = wv.x;
    b.y = wv.y;
    c = wmma_f32(a, b, c);
  }
  const int rbase = (lane < 16) ? 0 : 8;
  for (int r = 0; r < 8; ++r)
    Y[(size_t)(rbase + r) * Nout + n0 + m] = c[r];
}

// ---------------------------------------------------------------------------
// Flash-decode partial: one workgroup per (cache block, b, kv-head).
// K/V tiles (256x64 f32) land in padded LDS via the Tensor Data Mover,
// S = qK^T via WMMA, block-local softmax (m, l), O_partial = P*V via WMMA.
// 2048 workgroups keep HBM saturated.
// ---------------------------------------------------------------------------
__global__ __launch_bounds__(256) void attn_block(
    const float* __restrict__ blocks_k, const float* __restrict__ blocks_v,
    const float* __restrict__ q, const int* __restrict__ last_off_p,
    float* __restrict__ pm, float* __restrict__ pl, float* __restrict__ po) {
  __shared__ float Ks[TBLK * 68];   // stride 68 -> bank-conflict-free frags
  __shared__ float Vs[TBLK * 68];
  __shared__ float Qs[16 * 68];     // 4 real rows, zero-padded to 16
  __shared__ float Ss[16 * 260];    // scores/probs, rows 4..15 zero

  const int ib = blockIdx.x;                   // KV cache block
  const int bh = blockIdx.y;                   // b * NKV + h
  const int b = bh >> 3, h = bh & 7;
  const int tid = threadIdx.x;
  const int wave = tid >> 5, lane = tid & 31;
  const int nvalid = (ib == NBLK - 1) ? *last_off_p : TBLK;

  const size_t tbase = (((size_t)ib * BATCH + b) * NKV + h) * (TBLK * HD);

#if USE_TDM
  // --- DMA K/V tiles into padded LDS via Tensor Data Mover (wave 0) ---
  if (wave == 0) {
    tdm_load_tile(blocks_k + tbase, Ks, TBLK);
    tdm_load_tile(blocks_v + tbase, Vs, TBLK);
  }
#else
  const float4* Kg = (const float4*)(blocks_k + tbase);
  const float4* Vg = (const float4*)(blocks_v + tbase);
  for (int i = tid; i < TBLK * 16; i += 256) {         // 16 float4 per row
    const int row = i >> 4, c4 = i & 15;
    *(float4*)&Ks[row * 68 + (c4 << 2)] = Kg[i];
    *(float4*)&Vs[row * 68 + (c4 << 2)] = Vg[i];
  }
#endif

  // --- stage q (4x64), zero-pad rows 4..15; zero prob rows 4..15 ---
  const float* qrow = q + (size_t)b * EMBED + h * (G * HD);
  for (int i = tid; i < 16 * 68; i += 256) {
    const int row = i / 68, col = i - row * 68;
    Qs[i] = (row < G && col < HD) ? qrow[row * HD + col] : 0.0f;
  }
  for (int i = tid; i < 12 * 260; i += 256) Ss[G * 260 + i] = 0.0f;

#if USE_TDM
  if (wave == 0) tdm_wait();       // own TENSORcnt drained before barrier
#endif
  __syncthreads();

  const int m = lane & 15;
  const int koff = (lane < 16) ? 0 : 2;

  // --- S = q * K^T (each wave: two 16-key tiles, K-dim = 64) ---
  for (int t = 0; t < 2; ++t) {
    const int n0 = (wave * 2 + t) * 16;
    const float* arow = Qs + m * 68 + koff;
    const float* brow = Ks + (n0 + m) * 68 + koff;
    v8f c = {};
    for (int k0 = 0; k0 < HD; k0 += 4) {
      v2f a, bf;
      a.x = arow[k0];
      a.y = arow[k0 + 1];
      bf.x = brow[k0];
      bf.y = brow[k0 + 1];
      c = wmma_f32(a, bf, c);
    }
    if (lane < 16) {                       // rows 0..3 live in lanes 0..15
      const int key = n0 + lane;
      const bool valid = key < nvalid;
      for (int r = 0; r < G; ++r)
        Ss[r * 260 + key] = valid ? c[r] * SCALE : -3.0e38f;
    }
  }
  __syncthreads();

  // --- block-local softmax over 256 keys: waves 0..3 own rows 0..3 ---
  if (wave < G) {
    const int row = wave;
    float vals[8];
    float mx = -3.4e38f;
    for (int j = 0; j < 8; ++j) {
      vals[j] = Ss[row * 260 + lane * 8 + j];
      mx = fmaxf(mx, vals[j]);
    }
    for (int off = 16; off; off >>= 1) mx = fmaxf(mx, __shfl_xor(mx, off, 32));
    float s = 0.0f;
    for (int j = 0; j < 8; ++j) {
      const float p = __expf(vals[j] - mx);  // masked entries underflow to 0
      Ss[row * 260 + lane * 8 + j] = p;
      s += p;
    }
    for (int off = 16; off; off >>= 1) s += __shfl_xor(s, off, 32);
    if (lane == 0) {
      pm[((size_t)ib * BHN + bh) * G + row] = mx;
      pl[((size_t)ib * BHN + bh) * G + row] = s;
    }
  }
  __syncthreads();

  // --- O_partial = P * V (waves 0..3: one 16-wide d tile each, K = 256) ---
  if (wave < G) {
    const int n0 = wave * 16;
    const float* arow = Ss + m * 260 + koff;
    v8f c = {};
    for (int k0 = 0; k0 < TBLK; k0 += 4) {
      v2f a, bf;
      a.x = arow[k0];
      a.y = arow[k0 + 1];
      bf.x = Vs[(k0 + koff) * 68 + n0 + m];
      bf.y = Vs[(k0 + koff + 1) * 68 + n0 + m];
      c = wmma_f32(a, bf, c);
    }
    if (lane < 16) {
      float* orow = po + ((size_t)ib * BHN + bh) * (G * HD);
      for (int r = 0; r < G; ++r)
        orow[r * HD + n0 + lane] = c[r];
    }
  }
}

// ---------------------------------------------------------------------------
// Merge the 16 block partials (streaming softmax) + current-token K/V term.
// One block per (b, kv-head); threads = (g row, d) = 4 x 64.
// ---------------------------------------------------------------------------
__global__ __launch_bounds__(256) void attn_combine(
    const float* __restrict__ q, const float* __restrict__ kc,
    const float* __restrict__ vc, const float* __restrict__ pm,
    const float* __restrict__ pl, const float* __restrict__ po,
    float* __restrict__ atten) {
  const int bh = blockIdx.x;
  const int b = bh >> 3, h = bh & 7;
  const int tid = threadIdx.x;
  const int gi = tid >> 6, d = tid & 63;

  const float* qr = q + (size_t)b * EMBED + h * (G * HD) + gi * HD;
  const float* kr = kc + (size_t)b * (NKV * HD) + h * HD;
  const float* vr = vc + (size_t)b * (NKV * HD) + h * HD;

  float s_cur = 0.0f;
  for (int k = 0; k < HD; ++k) s_cur += qr[k] * kr[k];
  s_cur *= SCALE;

  float mg = s_cur;
  for (int ib = 0; ib < NBLK; ++ib)
    mg = fmaxf(mg, pm[((size_t)ib * BHN + bh) * G + gi]);

  const float ec = __expf(s_cur - mg);
  float acc = vr[d] * ec;
  float l = ec;
  for (int ib = 0; ib < NBLK; ++ib) {
    const size_t idx = ((size_t)ib * BHN + bh) * G + gi;
    const float w = __expf(pm[idx] - mg);
    l += pl[idx] * w;
    acc += po[idx * HD + d] * w;
  }
  atten[(size_t)b * EMBED + h * (G * HD) + gi * HD + d] = acc / l;
}

// ---------------------------------------------------------------------------
extern "C" void kernel_launch(void* const* d_in, const int* in_sizes, int n_in,
                              void* d_out, int out_size, void* d_ws,
                              size_t ws_size, hipStream_t stream) {
  const float* x  = (const float*)d_in[0];
  const float* bk = (const float*)d_in[1];
  const float* bv = (const float*)d_in[2];
  const float* Wq = (const float*)d_in[3];
  const float* Wk = (const float*)d_in[4];
  const float* Wv = (const float*)d_in[5];
  const float* Wo = (const float*)d_in[6];
  const int* last_off = (const int*)d_in[7];

  float* ws  = (float*)d_ws;          // workspace layout (floats):
  float* q   = ws;                    // 32768  : q  (16 x 2048)
  float* kc  = ws + 32768;            // 8192   : k_cur (16 x 512)
  float* vc  = ws + 40960;            // 8192   : v_cur (16 x 512)
  float* att = ws + 49152;            // 32768  : attended (16 x 2048)
  float* pm  = ws + 81920;            // 8192   : partial max  (16*128*4)
  float* pl  = ws + 90112;            // 8192   : partial sum
  float* po  = ws + 98304;            // 524288 : partial O (16*128*4*64)

  gemm16_xWt<<<16, 256, 0, stream>>>(x, Wq, q, EMBED);
  gemm16_xWt<<<4, 256, 0, stream>>>(x, Wk, kc, NKV * HD);
  gemm16_xWt<<<4, 256, 0, stream>>>(x, Wv, vc, NKV * HD);
  attn_block<<<dim3(NBLK, BHN), 256, 0, stream>>>(bk, bv, q, last_off, pm, pl, po);
  attn_combine<<<BHN, 256, 0, stream>>>(q, kc, vc, pm, pl, po, att);
  gemm16_xWt<<<16, 256, 0, stream>>>(att, Wo, (float*)d_out, EMBED);
}